// GraphConvolutionFirstOrder_53008486367985
// MI455X (gfx1250) — compile-verified
//
#include <hip/hip_runtime.h>
#include <hip/hip_bf16.h>

typedef __attribute__((ext_vector_type(2))) float v2f;
typedef __attribute__((ext_vector_type(8))) float v8f;

#define D 256            // D_IN == D_OUT == 256
#define TILES_N 4        // 4 x 16-wide N tiles per wave (64 cols)

__device__ __forceinline__ v8f wmma_f32(v2f a, v2f b, v8f c) {
    // (neg_a, A, neg_b, B, c_mod, C, reuse_a, reuse_b)
    return __builtin_amdgcn_wmma_f32_16x16x4_f32(false, a, false, b, (short)0, c,
                                                 false, false);
}

// One 128-thread block (4 waves) per 16-row M tile. Wave w covers columns
// [w*64, w*64+64). Each wave accumulates 8 C tiles: 4 for W_self, 4 for
// W_neighbor, sharing a single A fragment per K step (K step = 4, f32 WMMA).
__global__ void __launch_bounds__(128)
gcn_dual_gemm_kernel(const float* __restrict__ x,
                     const float* __restrict__ Wself,
                     const float* __restrict__ Wnbr,
                     const float* __restrict__ bias,
                     float* __restrict__ out,       // act_self + bias
                     float* __restrict__ support)   // x @ W_neighbor
{
    const int lane = threadIdx.x & 31;
    const int wave = threadIdx.x >> 5;      // 0..3 -> N quadrant
    const int g    = lane >> 4;             // lane group (0: lanes 0-15, 1: 16-31)
    const int ln   = lane & 15;             // lane within group

    const int mtile = blockIdx.x;           // 16-row tile index
    const int n0    = wave * (TILES_N * 16);

    // A fragment row for this lane (A 16x4 f32: M = lane%16, K = r + 2*g)
    const float* xrow = x + (size_t)(mtile * 16 + ln) * D;

    v8f cs[TILES_N] = {};  // self accumulators
    v8f cn[TILES_N] = {};  // neighbor accumulators

    for (int k = 0; k < D; k += 4) {
        // A: a[r] = x[m, k + 2*g + r]  (contiguous pair -> b64 load)
        const v2f a = *(const v2f*)(xrow + k + 2 * g);

        const int kb = (k + 2 * g) * D;     // row (k+2g) of W, row-major
#pragma unroll
        for (int t = 0; t < TILES_N; ++t) {
            const int n = n0 + t * 16 + ln; // B: N = lane%16 within tile
            v2f bs, bn;
            bs.x = Wself[kb + n];
            bs.y = Wself[kb + D + n];       // K = k + 2g + 1
            bn.x = Wnbr[kb + n];
            bn.y = Wnbr[kb + D + n];
            cs[t] = wmma_f32(a, bs, cs[t]);
            cn[t] = wmma_f32(a, bn, cn[t]);
        }
    }

    // C/D layout: VGPR r, lane group g -> row = r + 8*g, col = lane%16
#pragma unroll
    for (int t = 0; t < TILES_N; ++t) {
        const int col = n0 + t * 16 + ln;
        const float b = bias[col];
#pragma unroll
        for (int r = 0; r < 8; ++r) {
            const size_t row = (size_t)(mtile * 16 + r + 8 * g);
            out[row * D + col]     = cs[t][r] + b;
            support[row * D + col] = cn[t][r];
        }
    }
}

// One wave32 per edge: gather support[src] (2 x float4 per lane, coalesced
// L2-resident reads), scale by edge weight, scatter with relaxed agent-scope
// f32 atomics into the L2-resident output (global_atomic_add_f32, no return).
__global__ void __launch_bounds__(256)
gcn_edge_scatter_kernel(const float* __restrict__ support,
                        const int* __restrict__ edge_src,
                        const int* __restrict__ edge_dst,
                        const float* __restrict__ edge_weight,
                        float* __restrict__ out,
                        int n_edges)
{
    const int lane = threadIdx.x & 31;
    const int e = blockIdx.x * (blockDim.x >> 5) + (threadIdx.x >> 5);
    if (e >= n_edges) return;

    const int   s = edge_src[e];
    const int   d = edge_dst[e];
    const float w = edge_weight[e];

    const float4* __restrict__ srow = (const float4*)(support + (size_t)s * D);
    float* __restrict__ drow = out + (size_t)d * D;

#pragma unroll
    for (int i = 0; i < 2; ++i) {
        const int q = lane + i * 32;        // float4 index within the row
        const float4 v = srow[q];
        float* p = drow + q * 4;
        __hip_atomic_fetch_add(p + 0, v.x * w, __ATOMIC_RELAXED, __HIP_MEMORY_SCOPE_AGENT);
        __hip_atomic_fetch_add(p + 1, v.y * w, __ATOMIC_RELAXED, __HIP_MEMORY_SCOPE_AGENT);
        __hip_atomic_fetch_add(p + 2, v.z * w, __ATOMIC_RELAXED, __HIP_MEMORY_SCOPE_AGENT);
        __hip_atomic_fetch_add(p + 3, v.w * w, __ATOMIC_RELAXED, __HIP_MEMORY_SCOPE_AGENT);
    }
}

extern "C" void kernel_launch(void* const* d_in, const int* in_sizes, int n_in,
                              void* d_out, int out_size, void* d_ws, size_t ws_size,
                              hipStream_t stream) {
    const float* x      = (const float*)d_in[0];
    const int*   esrc   = (const int*)  d_in[1];
    const int*   edst   = (const int*)  d_in[2];
    const float* eweight= (const float*)d_in[3];
    const float* Wself  = (const float*)d_in[4];
    const float* Wnbr   = (const float*)d_in[5];
    const float* bias   = (const float*)d_in[6];

    float* out     = (float*)d_out;
    float* support = (float*)d_ws;     // [n_nodes, 256] scratch

    const int n_nodes = in_sizes[0] / D;   // 50000
    const int n_edges = in_sizes[1];       // 800000

    // GEMM phase: writes every element of out (act_self + bias) and support.
    const int mtiles = n_nodes / 16;       // 50000/16 = 3125 exactly
    gcn_dual_gemm_kernel<<<mtiles, 128, 0, stream>>>(x, Wself, Wnbr, bias,
                                                     out, support);

    // Scatter phase: one wave per edge, 8 waves per 256-thread block.
    const int waves_per_block = 256 / 32;
    const int blocks = (n_edges + waves_per_block - 1) / waves_per_block;
    gcn_edge_scatter_kernel<<<blocks, 256, 0, stream>>>(support, esrc, edst,
                                                        eweight, out, n_edges);
}